// OverlapWindow_89069031784895
// MI455X (gfx1250) — compile-verified
//
#include <hip/hip_runtime.h>

// Problem constants (from the reference)
#define N_CONTEXT 9
#define WINDOW    19              // 2*N_CONTEXT + 1
#define CH        26
#define ROWF      (WINDOW * CH)   // 494 floats per output time-step
#define ROWFIX    (ROWF - CH)     // 468 : tile-index fixup at row wrap

// Tiling
#define BT        64                          // time steps per block (even -> 16B-aligned f4 stores)
#define TILE_ROWS (BT + 2 * N_CONTEXT)        // 82 input rows staged
#define TILE_F    (TILE_ROWS * CH)            // 2132 floats = 8528 B of LDS
#define BLOCK     256                         // 8 wave32 waves

// Native clang vector type: required by __builtin_nontemporal_store
typedef float __attribute__((ext_vector_type(4))) floatx4;

// ---- CDNA5 async global->LDS path (confirmed present by earlier rounds) ----
#define HAVE_ASYNC_LDS 0
#define HAVE_WAIT_ASYNC 0
#if defined(__has_builtin)
#  if __has_builtin(__builtin_amdgcn_global_load_async_to_lds_b32)
#    undef HAVE_ASYNC_LDS
#    define HAVE_ASYNC_LDS 1
#  endif
#  if __has_builtin(__builtin_amdgcn_s_wait_asynccnt)
#    undef HAVE_WAIT_ASYNC
#    define HAVE_WAIT_ASYNC 1
#  endif
#endif

// Parameter types per clang's diagnostic: (as1 int*, as3 int*, i32 offset, i32 cpol)
typedef __attribute__((address_space(1))) int async_g_i32;
typedef __attribute__((address_space(3))) int async_l_i32;

__global__ __launch_bounds__(BLOCK)
void OverlapWindow_kernel(const float* __restrict__ x,
                          float* __restrict__ out,
                          int T) {
    __shared__ float tile[TILE_F];

    const int t0    = (int)blockIdx.x * BT;
    const int gbase = (t0 - N_CONTEXT) * CH;   // may be negative (left padding)
    const int limit = T * CH;

    // ---- Stage: contiguous x[(t0-9)*26 .. (t0+BT+9)*26) -> LDS via async copies ----
    for (int i = (int)threadIdx.x; i < TILE_F; i += BLOCK) {
        const int gf = gbase + i;
        if (gf >= 0 && gf < limit) {
            // GLOBAL_LOAD_ASYNC_TO_LDS_B32: per-lane global -> per-lane LDS, ASYNCcnt-tracked.
#if HAVE_ASYNC_LDS
            __builtin_amdgcn_global_load_async_to_lds_b32(
                (async_g_i32*)(x + gf), (async_l_i32*)(tile + i), 0, 0);
#else
            tile[i] = x[gf];
#endif
        } else {
            tile[i] = 0.0f;   // zero padding region
        }
    }

#if HAVE_ASYNC_LDS
    // Each wave drains its own async copies; barrier covers cross-wave visibility.
#  if HAVE_WAIT_ASYNC
    __builtin_amdgcn_s_wait_asynccnt(0);
#  else
    asm volatile("s_wait_asynccnt 0" ::: "memory");
#  endif
#endif
    __syncthreads();

    // ---- Writeback: dense float4 (b128) non-temporal stores ----
    // Key identity: out region offset r = tl*494 + j  reads  tile[tl*26 + j]
    // (since j = w*26 + c  =>  (tl+w)*26 + c = tl*26 + j).  All index math is
    // incremental adds with a single conditional -468 fixup at row wraps.
    const int nT = (T - t0 < BT) ? (T - t0) : BT;
    if (nT <= 0) return;
    const int nF = nT * ROWF;        // floats in this block's output region
    const int n4 = nF >> 2;          // nT even for all blocks here -> nF % 4 == 0

    // Region start t0*494 floats; t0 multiple of 64 -> 16B aligned.
    floatx4* __restrict__ out4 = reinterpret_cast<floatx4*>(out + (size_t)t0 * ROWF);

    {
        int i   = (int)threadIdx.x;  // float4 index within region
        int r   = i << 2;            // float offset (element 0)
        int tl  = r / ROWF;          // one-time small division (r < 1024)
        int j   = r - tl * ROWF;     // position within the 494-float output row
        int idx = tl * CH + j;       // tile float index for element 0

        for (; i < n4; i += BLOCK) {
            floatx4 v;
            int jj = j, ii = idx;
#pragma unroll
            for (int e = 0; e < 4; ++e) {
                v[e] = tile[ii];
                ++jj; ++ii;
                if (jj == ROWF) { jj = 0; ii -= ROWFIX; }  // row wrap inside float4
            }
            __builtin_nontemporal_store(v, &out4[i]);      // write-once stream, NT hint

            // Advance BLOCK float4s = 1024 floats = 2 rows + 36
            j   += 1024 - 2 * ROWF;            // +36
            idx += 2 * CH + (1024 - 2 * ROWF); // +88
            if (j >= ROWF) { j -= ROWF; idx -= ROWFIX; }   // at most one wrap (493+36<988)
        }
    }

    // Scalar tail (only if nF % 4 != 0; dead for this problem size, kept for robustness).
    for (int r = (n4 << 2) + (int)threadIdx.x; r < nF; r += BLOCK) {
        const int tl = r / ROWF;
        const int j  = r - tl * ROWF;
        out[(size_t)t0 * ROWF + r] = tile[tl * CH + j];
    }
}

extern "C" void kernel_launch(void* const* d_in, const int* in_sizes, int n_in,
                              void* d_out, int out_size, void* d_ws, size_t ws_size,
                              hipStream_t stream) {
    const float* x  = (const float*)d_in[0];
    float* out      = (float*)d_out;
    const int T     = in_sizes[0] / CH;          // 500000
    const int grid  = (T + BT - 1) / BT;         // 7813 blocks
    OverlapWindow_kernel<<<dim3(grid), dim3(BLOCK), 0, stream>>>(x, out, T);
}